// GCN_48378511622512
// MI455X (gfx1250) — compile-verified
//
#include <hip/hip_runtime.h>

typedef __attribute__((ext_vector_type(2))) float v2f;
typedef __attribute__((ext_vector_type(8))) float v8f;

#define IN_DIM 512
#define HID    64
#define OUTD   32

// ---------------- degree / normalization ----------------

__global__ __launch_bounds__(256) void k_deg_init(float* __restrict__ deg, int n) {
    int i = blockIdx.x * blockDim.x + threadIdx.x;
    if (i < n) deg[i] = 1.0f;  // self loop contributes 1 to every node's degree
}

__global__ __launch_bounds__(256) void k_deg_count(const int* __restrict__ dst,
                                                   float* __restrict__ deg, int E) {
    int i = blockIdx.x * blockDim.x + threadIdx.x;
    if (i < E) {
        (void)__hip_atomic_fetch_add(&deg[dst[i]], 1.0f,
                                     __ATOMIC_RELAXED, __HIP_MEMORY_SCOPE_AGENT);
    }
}

__global__ __launch_bounds__(256) void k_dinv(float* __restrict__ deg, int n) {
    int i = blockIdx.x * blockDim.x + threadIdx.x;
    if (i < n) {
        float d = deg[i];
        deg[i] = (d > 0.0f) ? rsqrtf(d) : 0.0f;
    }
}

// ---------------- weight repack: Wp[k/2][n] = {W[k][n], W[k+1][n]} ----------------
// Makes each WMMA B-operand fetch a single 8-byte load.

template <int NOUT>
__global__ __launch_bounds__(256) void k_packW(const float* __restrict__ W,
                                               v2f* __restrict__ Wp, int K) {
    int i = blockIdx.x * blockDim.x + threadIdx.x;
    int total = (K / 2) * NOUT;
    if (i >= total) return;
    int k2 = i / NOUT;
    int n  = i % NOUT;
    v2f p;
    p.x = W[(size_t)(2 * k2) * NOUT + n];
    p.y = W[(size_t)(2 * k2 + 1) * NOUT + n];
    Wp[i] = p;
}

// ---------------- WMMA GEMM: out[row] = (X[row] @ W) * dinv[row] ----------------
// One wave per 32-row tile (2 M-tiles) to amortize B loads across 2 WMMAs.
// V_WMMA_F32_16X16X4_F32 layouts:
//   A (16x4, 2 VGPRs): lane L: m=L&15, khh=L>>4; vgpr0=A[m][kk+2*khh], vgpr1=A[m][kk+2*khh+1]
//   B (4x16, 2 VGPRs): mirrors A: vgpr0=B[kk+2*khh][n], vgpr1=B[kk+2*khh+1][n], n=L&15
//   C/D (16x16, 8 VGPRs): vgpr r = C[r + 8*khh][n]

template <int K, int NOUT>
__global__ __launch_bounds__(256) void k_gemm_wmma(const float* __restrict__ X,
                                                   const v2f* __restrict__ Wp,
                                                   const float* __restrict__ W,
                                                   const float* __restrict__ dinv,
                                                   float* __restrict__ out, int nrows) {
    constexpr int NT = NOUT / 16;
    constexpr int MT = 2;  // M-tiles per wave
    int wave = (int)((blockIdx.x * (unsigned)blockDim.x + threadIdx.x) >> 5);
    int lane = threadIdx.x & 31;
    int row0 = wave * (16 * MT);
    if (row0 >= nrows) return;

    int m   = lane & 15;
    int khh = lane >> 4;  // 0 or 1

    if (row0 + 16 * MT > nrows) {
        // Scalar fallback for a partial tile (not hit for N=100000: 32 | N).
        for (int row = row0; row < nrows; ++row) {
            const float* xr = X + (size_t)row * K;
            for (int col = lane; col < NOUT; col += 32) {
                float s = 0.0f;
                for (int k = 0; k < K; ++k) s += xr[k] * W[(size_t)k * NOUT + col];
                out[(size_t)row * NOUT + col] = s * dinv[row];
            }
        }
        return;
    }

    v8f acc[MT][NT] = {};
    const float* xrow0 = X + (size_t)(row0 + m) * K;
    const float* xrow1 = xrow0 + (size_t)16 * K;

    for (int kk = 0; kk < K; kk += 4) {
        int ka = kk + 2 * khh;
        v2f a0 = *(const v2f*)(xrow0 + ka);           // b64, 8B-aligned (ka even)
        v2f a1 = *(const v2f*)(xrow1 + ka);
        const v2f* wrow = Wp + (size_t)((kk >> 1) + khh) * NOUT;
#pragma unroll
        for (int nt = 0; nt < NT; ++nt) {
            v2f b = wrow[nt * 16 + m];                // single b64
            acc[0][nt] = __builtin_amdgcn_wmma_f32_16x16x4_f32(
                false, a0, false, b, (short)0, acc[0][nt], false, false);
            acc[1][nt] = __builtin_amdgcn_wmma_f32_16x16x4_f32(
                false, a1, false, b, (short)0, acc[1][nt], false, false);
        }
    }

#pragma unroll
    for (int mt = 0; mt < MT; ++mt) {
        int rbase = row0 + mt * 16 + khh * 8;
#pragma unroll
        for (int nt = 0; nt < NT; ++nt) {
#pragma unroll
            for (int r = 0; r < 8; ++r) {
                int row = rbase + r;
                out[(size_t)row * NOUT + nt * 16 + m] = acc[mt][nt][r] * dinv[row];
            }
        }
    }
}

// ---------------- edge scatter-add (L2-resident float atomics) ----------------
// agg[dst] += table[src]; 32 lanes share one edge, consecutive lanes = consecutive
// channels -> coalesced gather, atomics land in L2 (both arrays << 192MB).

__global__ __launch_bounds__(256) void k_scatter64(const int* __restrict__ src,
                                                   const int* __restrict__ dst,
                                                   const float* __restrict__ tab,
                                                   float* __restrict__ agg, int E) {
    int t = blockIdx.x * blockDim.x + threadIdx.x;
    int e = t >> 5;
    if (e >= E) return;
    int cpair = t & 31;  // 2 channels per lane
    int s = src[e], d = dst[e];
    v2f v = ((const v2f*)(tab + (size_t)s * 64))[cpair];
    float* op = agg + (size_t)d * 64 + cpair * 2;
    (void)__hip_atomic_fetch_add(op,     v.x, __ATOMIC_RELAXED, __HIP_MEMORY_SCOPE_AGENT);
    (void)__hip_atomic_fetch_add(op + 1, v.y, __ATOMIC_RELAXED, __HIP_MEMORY_SCOPE_AGENT);
}

__global__ __launch_bounds__(256) void k_scatter32(const int* __restrict__ src,
                                                   const int* __restrict__ dst,
                                                   const float* __restrict__ tab,
                                                   float* __restrict__ agg, int E) {
    int t = blockIdx.x * blockDim.x + threadIdx.x;
    int e = t >> 5;
    if (e >= E) return;
    int c = t & 31;
    int s = src[e], d = dst[e];
    float v = tab[(size_t)s * 32 + c];
    (void)__hip_atomic_fetch_add(agg + (size_t)d * 32 + c, v,
                                 __ATOMIC_RELAXED, __HIP_MEMORY_SCOPE_AGENT);
}

// ---------------- finalize: h = [relu](h * dinv[row] + bias[c]) ----------------

template <int C, bool RELU>
__global__ __launch_bounds__(256) void k_finalize(float* __restrict__ h,
                                                  const float* __restrict__ dinv,
                                                  const float* __restrict__ bias,
                                                  int total) {
    int i = blockIdx.x * blockDim.x + threadIdx.x;
    if (i >= total) return;
    int row = i / C;
    int c   = i % C;
    float v = h[i] * dinv[row] + bias[c];
    if (RELU) v = fmaxf(v, 0.0f);
    h[i] = v;
}

// ---------------- driver ----------------

extern "C" void kernel_launch(void* const* d_in, const int* in_sizes, int n_in,
                              void* d_out, int out_size, void* d_ws, size_t ws_size,
                              hipStream_t stream) {
    const float* x  = (const float*)d_in[0];
    const int*   ei = (const int*)d_in[1];
    const float* W1 = (const float*)d_in[2];
    const float* b1 = (const float*)d_in[3];
    const float* W2 = (const float*)d_in[4];
    const float* b2 = (const float*)d_in[5];

    const int N = in_sizes[0] / IN_DIM;
    const int E = in_sizes[1] / 2;
    const int* src = ei;
    const int* dst = ei + E;
    float* out = (float*)d_out;

    // workspace: dinv [N] | bufA [N*HID] | bufB [N*HID] | Wp1 | Wp2
    float* dinv = (float*)d_ws;
    float* bufA = dinv + N;
    float* bufB = bufA + (size_t)N * HID;
    v2f*   Wp1  = (v2f*)(bufB + (size_t)N * HID);                 // (IN_DIM/2)*HID v2f
    v2f*   Wp2  = Wp1 + (size_t)(IN_DIM / 2) * HID;               // (HID/2)*OUTD v2f

    const int B = 256;
    const int nwaves = (N + 31) / 32;  // 32 rows per wave
    const unsigned gemm_blocks = (unsigned)(((size_t)nwaves * 32 + B - 1) / B);
    const unsigned edge_blocks = (unsigned)(((size_t)E * 32 + B - 1) / B);

    // normalization: deg -> dinv (in place)
    k_deg_init<<<(N + B - 1) / B, B, 0, stream>>>(dinv, N);
    k_deg_count<<<(E + B - 1) / B, B, 0, stream>>>(dst, dinv, E);
    k_dinv<<<(N + B - 1) / B, B, 0, stream>>>(dinv, N);

    // repack weights into k-paired float2 layout
    k_packW<HID><<<((IN_DIM / 2) * HID + B - 1) / B, B, 0, stream>>>(W1, Wp1, IN_DIM);
    k_packW<OUTD><<<((HID / 2) * OUTD + B - 1) / B, B, 0, stream>>>(W2, Wp2, HID);

    // ---- layer 1: h1 = relu( D^-1/2 (A+I) D^-1/2 x W1 + b1 ) ----
    k_gemm_wmma<IN_DIM, HID><<<gemm_blocks, B, 0, stream>>>(x, Wp1, W1, dinv, bufA, N);
    // seed aggregation with the self-loop term (agg = scaled_h)
    hipMemcpyAsync(bufB, bufA, (size_t)N * HID * sizeof(float),
                   hipMemcpyDeviceToDevice, stream);
    k_scatter64<<<edge_blocks, B, 0, stream>>>(src, dst, bufA, bufB, E);
    k_finalize<HID, true><<<((size_t)N * HID + B - 1) / B, B, 0, stream>>>(
        bufB, dinv, b1, N * HID);

    // ---- layer 2: out = D^-1/2 (A+I) D^-1/2 h1 W2 + b2 ----
    k_gemm_wmma<HID, OUTD><<<gemm_blocks, B, 0, stream>>>(bufB, Wp2, W2, dinv, bufA, N);
    hipMemcpyAsync(out, bufA, (size_t)N * OUTD * sizeof(float),
                   hipMemcpyDeviceToDevice, stream);
    k_scatter32<<<edge_blocks, B, 0, stream>>>(src, dst, bufA, out, E);
    k_finalize<OUTD, false><<<((size_t)N * OUTD + B - 1) / B, B, 0, stream>>>(
        out, dinv, b2, N * OUTD);
}